// CRFasRNN_40355512713232
// MI455X (gfx1250) — compile-verified
//
#include <hip/hip_runtime.h>
#include <cmath>

typedef __attribute__((ext_vector_type(2))) float v2f;
typedef __attribute__((ext_vector_type(8))) float v8f;

#define BATCH 2
#define NCH   21
#define CPAD  24          // padded channel count (multiple of WMMA K=4)
#define HH    256
#define WW    256
#define HWSZ  (HH*WW)
#define K2SZ  25
#define NITER 5

struct K25 { float v[25]; };

// ---------------------------------------------------------------------------
// q = softmax(logits) over channel dim. One thread per (b,pixel).
// ---------------------------------------------------------------------------
__global__ __launch_bounds__(256)
void softmax_kernel(const float* __restrict__ logits, float* __restrict__ q) {
    int idx = blockIdx.x * blockDim.x + threadIdx.x;   // over B*HW
    if (idx >= BATCH * HWSZ) return;
    int p = idx % HWSZ, b = idx / HWSZ;
    const float* lg = logits + (size_t)b * NCH * HWSZ + p;
    float v[NCH];
    float mx = -3.4e38f;
    #pragma unroll
    for (int c = 0; c < NCH; ++c) { v[c] = lg[c * HWSZ]; mx = fmaxf(mx, v[c]); }
    float sum = 0.f;
    #pragma unroll
    for (int c = 0; c < NCH; ++c) { v[c] = expf(v[c] - mx); sum += v[c]; }
    float inv = 1.f / sum;
    float* qo = q + (size_t)b * NCH * HWSZ + p;
    #pragma unroll
    for (int c = 0; c < NCH; ++c) qo[c * HWSZ] = v[c] * inv;
}

// ---------------------------------------------------------------------------
// One-time prep:
//   negA[32][24] = -compat (zero-padded)  -> unguarded WMMA A loads; the
//     pre-negation lets the accumulator start at logits so the WMMA chain
//     directly yields D = logits - compat@s (f32 WMMA has no A/B negate).
//   Zero the pad channels (21..23) of the message buffer s.
// ---------------------------------------------------------------------------
__global__ __launch_bounds__(256)
void prep_pad_kernel(const float* __restrict__ compat, float* __restrict__ negA,
                     float* __restrict__ s) {
    int t = blockIdx.x * blockDim.x + threadIdx.x;
    if (t < 32 * CPAD) {
        int m = t / CPAD, j = t % CPAD;
        negA[t] = (m < NCH && j < NCH) ? -compat[m * NCH + j] : 0.f;
    }
    if (t < BATCH * (CPAD - NCH) * HWSZ) {
        int p = t % HWSZ, bc = t / HWSZ;
        int b = bc / (CPAD - NCH), c = NCH + bc % (CPAD - NCH);
        s[((size_t)b * CPAD + c) * HWSZ + p] = 0.f;
    }
}

// ---------------------------------------------------------------------------
// cw[b,k2,p] = gk[k2] + sw[k2]*exp(-|img(nbr)-img(ctr)|^2 / (2*0.1^2))
// ---------------------------------------------------------------------------
__global__ __launch_bounds__(256)
void weight_kernel(const float* __restrict__ img, float* __restrict__ cw,
                   K25 gk, K25 sw) {
    int idx = blockIdx.x * blockDim.x + threadIdx.x;   // over B*HW
    if (idx >= BATCH * HWSZ) return;
    int p = idx % HWSZ, b = idx / HWSZ;
    int y = p >> 8, x = p & 255;
    const float* im = img + (size_t)b * 3 * HWSZ;
    float r0 = im[p], g0 = im[HWSZ + p], b0 = im[2 * HWSZ + p];
    float* out = cw + (size_t)b * K2SZ * HWSZ + p;
    #pragma unroll
    for (int i = 0; i < 5; ++i) {
        int yy = y + i - 2;
        #pragma unroll
        for (int j = 0; j < 5; ++j) {
            int xx = x + j - 2;
            float rn = 0.f, gn = 0.f, bn = 0.f;
            if (yy >= 0 && yy < HH && xx >= 0 && xx < WW) {
                int pn = yy * WW + xx;
                rn = im[pn]; gn = im[HWSZ + pn]; bn = im[2 * HWSZ + pn];
            }
            float d = (rn - r0) * (rn - r0) + (gn - g0) * (gn - g0)
                    + (bn - b0) * (bn - b0);
            int k2 = i * 5 + j;
            out[k2 * HWSZ] = gk.v[k2] + sw.v[k2] * expf(-d * 50.0f);
        }
    }
}

// ---------------------------------------------------------------------------
// s[b,c,p] = sum_k2 q[b,c,nbr(k2)] * cw[b,k2,p]   (zero padding: q OOB = 0)
// s has CPAD channels per batch; channels 21..23 stay zero.
// ---------------------------------------------------------------------------
__global__ __launch_bounds__(256)
void message_kernel(const float* __restrict__ q, const float* __restrict__ cw,
                    float* __restrict__ s) {
    int idx = blockIdx.x * blockDim.x + threadIdx.x;   // over B*C*HW
    if (idx >= BATCH * NCH * HWSZ) return;
    int p = idx % HWSZ, bc = idx / HWSZ;
    int b = bc / NCH, c = bc % NCH;
    int y = p >> 8, x = p & 255;
    const float* qc = q + (size_t)bc * HWSZ;
    const float* wb = cw + (size_t)b * K2SZ * HWSZ + p;
    float acc = 0.f;
    #pragma unroll
    for (int i = 0; i < 5; ++i) {
        int yy = y + i - 2;
        #pragma unroll
        for (int j = 0; j < 5; ++j) {
            int xx = x + j - 2;
            if (yy >= 0 && yy < HH && xx >= 0 && xx < WW) {
                acc += qc[yy * WW + xx] * wb[(i * 5 + j) * HWSZ];
            }
        }
    }
    s[((size_t)b * CPAD + c) * HWSZ + p] = acc;
}

// ---------------------------------------------------------------------------
// D = logits + (-compat)@s via V_WMMA_F32_16X16X4_F32 (acc seeded with
// logits in D-layout), fused with q = softmax(D). One wave per 16 pixels.
// All A/B loads unguarded (padded buffers) -> clean EXEC-full MAC loop.
// ---------------------------------------------------------------------------
__global__ __launch_bounds__(256)
void compat_softmax_wmma(const float* __restrict__ s,
                         const float* __restrict__ logits,
                         const float* __restrict__ negA,   // [32][24]
                         float* __restrict__ qout) {
    const int lane = threadIdx.x & 31;
    const int wave = threadIdx.x >> 5;
    const int tile = blockIdx.x * 8 + wave;    // 16-pixel tile id over B*HW/16
    const int pix0 = tile * 16;
    const int b    = pix0 / HWSZ;
    const int n    = lane & 15;                // pixel-in-tile / matrix row
    const int half = lane >> 4;
    const int kb   = half * 2;                 // K-pair base for this lane
    const int p    = (pix0 - b * HWSZ) + n;    // pixel within batch
    const float* sB = s + (size_t)b * CPAD * HWSZ + p;
    const float* lg = logits + (size_t)b * NCH * HWSZ + p;

    // Preload entire 32x24 (-compat) matrix into wave registers.
    v2f a0[6], a1[6];
    #pragma unroll
    for (int kc = 0; kc < 6; ++kc) {
        const int j = kc * 4 + kb;
        a0[kc].x = negA[n * CPAD + j];
        a0[kc].y = negA[n * CPAD + j + 1];
        a1[kc].x = negA[(16 + n) * CPAD + j];
        a1[kc].y = negA[(16 + n) * CPAD + j + 1];
    }

    // Seed accumulators with logits in D layout:
    //   acc0[r] -> channel 8*half+r, acc1[r] -> channel 16+8*half+r.
    v8f acc0, acc1;
    #pragma unroll
    for (int r = 0; r < 8; ++r) {
        acc0[r] = lg[(8 * half + r) * HWSZ];
        int c = 16 + 8 * half + r;
        acc1[r] = (c < NCH) ? lg[c * HWSZ] : 0.f;
    }

    #pragma unroll
    for (int kc = 0; kc < 6; ++kc) {
        const int j = kc * 4 + kb;
        v2f bv;
        bv.x = sB[j * HWSZ];
        bv.y = sB[(j + 1) * HWSZ];
        acc0 = __builtin_amdgcn_wmma_f32_16x16x4_f32(
                   false, a0[kc], false, bv, (short)0, acc0, false, false);
        acc1 = __builtin_amdgcn_wmma_f32_16x16x4_f32(
                   false, a1[kc], false, bv, (short)0, acc1, false, false);
    }

    // Fused softmax over 21 channels: one cross-half shuffle for max & sum.
    float e0[8], e1[8];
    float mx = -3.4e38f;
    #pragma unroll
    for (int r = 0; r < 8; ++r) { e0[r] = acc0[r]; mx = fmaxf(mx, e0[r]); }
    #pragma unroll
    for (int r = 0; r < 8; ++r) {
        int c = 16 + 8 * half + r;
        e1[r] = (c < NCH) ? acc1[r] : -3.4e38f;
        mx = fmaxf(mx, e1[r]);
    }
    mx = fmaxf(mx, __shfl_xor(mx, 16, 32));
    float sum = 0.f;
    #pragma unroll
    for (int r = 0; r < 8; ++r) { e0[r] = expf(e0[r] - mx); sum += e0[r]; }
    #pragma unroll
    for (int r = 0; r < 8; ++r) {
        int c = 16 + 8 * half + r;
        e1[r] = (c < NCH) ? expf(e1[r] - mx) : 0.f;
        sum += e1[r];
    }
    sum += __shfl_xor(sum, 16, 32);
    float inv = 1.f / sum;

    float* qo = qout + (size_t)b * NCH * HWSZ + p;
    #pragma unroll
    for (int r = 0; r < 8; ++r) qo[(8 * half + r) * HWSZ] = e0[r] * inv;
    #pragma unroll
    for (int r = 0; r < 8; ++r) {
        int c = 16 + 8 * half + r;
        if (c < NCH) qo[c * HWSZ] = e1[r] * inv;
    }
}

// ---------------------------------------------------------------------------
static void gauss25(float sigma, K25* out) {
    float sum = 0.f;
    for (int i = 0; i < 5; ++i)
        for (int j = 0; j < 5; ++j) {
            float dy = (float)i - 2.0f, dx = (float)j - 2.0f;
            float v = expf(-(dx * dx + dy * dy) / (2.0f * sigma * sigma));
            out->v[i * 5 + j] = v; sum += v;
        }
    for (int k = 0; k < 25; ++k) out->v[k] /= sum;
}

extern "C" void kernel_launch(void* const* d_in, const int* in_sizes, int n_in,
                              void* d_out, int out_size, void* d_ws, size_t ws_size,
                              hipStream_t stream) {
    (void)in_sizes; (void)n_in; (void)out_size; (void)ws_size;
    const float* logits = (const float*)d_in[0];   // [B,21,256,256]
    const float* img    = (const float*)d_in[1];   // [B,3,256,256]
    // d_in[2] = gt_edges: dead code in reference, unused
    const float* compat = (const float*)d_in[3];   // [21,21]

    float* qbuf  = (float*)d_ws;                          // B*21*HW
    float* sbuf  = qbuf + (size_t)BATCH * NCH * HWSZ;     // B*24*HW (padded)
    float* cwbuf = sbuf + (size_t)BATCH * CPAD * HWSZ;    // B*25*HW
    float* negA  = cwbuf + (size_t)BATCH * K2SZ * HWSZ;   // 32*24

    K25 gk, sw;
    gauss25(0.1f, &gk);   // GKS=5, GSIG=0.1 (spatial conv kernel)
    gauss25(5.0f, &sw);   // BKS=5, BSSIG=5.0 (bilateral spatial term)

    const int nPix  = BATCH * HWSZ;                    // 131072
    const int nBCP  = BATCH * NCH * HWSZ;              // 2752512
    const int nPrep = BATCH * (CPAD - NCH) * HWSZ;     // 393216 (covers 32*24 too)

    softmax_kernel<<<(nPix + 255) / 256, 256, 0, stream>>>(logits, qbuf);
    prep_pad_kernel<<<(nPrep + 255) / 256, 256, 0, stream>>>(compat, negA, sbuf);
    weight_kernel<<<(nPix + 255) / 256, 256, 0, stream>>>(img, cwbuf, gk, sw);

    for (int it = 0; it < NITER; ++it) {
        message_kernel<<<(nBCP + 255) / 256, 256, 0, stream>>>(qbuf, cwbuf, sbuf);
        float* qdst = (it == NITER - 1) ? (float*)d_out : qbuf;
        // B*HW pixels / (8 waves * 16 pixels) = 1024 blocks, exact
        compat_softmax_wmma<<<nPix / 128, 256, 0, stream>>>(sbuf, logits, negA, qdst);
    }
}